// TonerTuClosure_11862699671596
// MI455X (gfx1250) — compile-verified
//
#include <hip/hip_runtime.h>
#include <cmath>

#define NX 512
#define NY 512
#define NC 3
#define TX 64          // interior tile width  (one thread per point)
#define TY 8           // interior tile height
#define BLOCK 512      // 16 wave32s
#define LXW (TX + 2)   // 66 (with halo)
#define LYH (TY + 2)   // 10
#define DT_STEP 0.05f

#if defined(__HIP_DEVICE_COMPILE__) && defined(__gfx1250__) && \
    __has_builtin(__builtin_amdgcn_global_load_async_to_lds_b32) && \
    __has_builtin(__builtin_amdgcn_s_wait_asynccnt)
#define USE_ASYNC_LDS 1
typedef __attribute__((address_space(1))) int async_gint_t;   // global
typedef __attribute__((address_space(3))) int async_lint_t;   // LDS
#else
#define USE_ASYNC_LDS 0
#endif

__global__ __launch_bounds__(BLOCK) void toner_tu_step_kernel(
    const float* __restrict__ state,
    const float* __restrict__ p_v0,
    const float* __restrict__ p_Drho,
    const float* __restrict__ p_alpha,
    const float* __restrict__ p_beta,
    const float* __restrict__ p_DP,
    float* __restrict__ out_pred,
    float* __restrict__ out_dt)
{
    __shared__ float tile[NC][LYH][LXW];   // 3*10*66*4 = 7920 B

    const int tid = threadIdx.x;
    const int x0  = blockIdx.x * TX;
    const int y0  = blockIdx.y * TY;
    const int b   = blockIdx.z;

    const size_t plane = (size_t)NX * NY;
    const size_t batch = (size_t)NC * plane;
    const float* base  = state + (size_t)b * batch;

    // ---- cooperative halo-tile load, one channel at a time ----
    // per channel: LYH*LXW = 660 elements; async DMA global -> LDS
    for (int c = 0; c < NC; ++c) {
        const float* cbase = base + (size_t)c * plane;
        for (int i = tid; i < LYH * LXW; i += BLOCK) {
            const int ly = i / LXW;
            const int lx = i - ly * LXW;
            const int gx = (x0 + lx - 1) & (NX - 1);   // periodic wrap
            const int gy = (y0 + ly - 1) & (NY - 1);
            const float* gp = cbase + (size_t)gy * NX + gx;
#if USE_ASYNC_LDS
            // CDNA5 async DMA: global -> LDS, no VGPR staging (ASYNCcnt)
            __builtin_amdgcn_global_load_async_to_lds_b32(
                (async_gint_t*)gp,
                (async_lint_t*)&tile[c][ly][lx],
                0, 0);
#else
            tile[c][ly][lx] = *gp;
#endif
        }
    }
#if USE_ASYNC_LDS
    __builtin_amdgcn_s_wait_asynccnt(0);   // my wave's async loads landed in LDS
#endif
    __syncthreads();                       // all waves' tiles visible

    // ---- uniform scalar parameters ----
    const float v0    = *p_v0;
    const float Drho  = *p_Drho;
    const float alpha = *p_alpha;
    const float beta  = *p_beta;
    const float DP    = *p_DP;

    // ---- per-thread stencil point ----
    const int lx = (tid & (TX - 1)) + 1;
    const int ly = (tid >> 6) + 1;
    const int gx = x0 + lx - 1;
    const int gy = y0 + ly - 1;

    const float rc   = tile[0][ly][lx];
    const float rxm  = tile[0][ly][lx - 1], rxp = tile[0][ly][lx + 1];
    const float rym  = tile[0][ly - 1][lx], ryp = tile[0][ly + 1][lx];
    const float pxc  = tile[1][ly][lx];
    const float pxxm = tile[1][ly][lx - 1], pxxp = tile[1][ly][lx + 1];
    const float pxym = tile[1][ly - 1][lx], pxyp = tile[1][ly + 1][lx];
    const float pyc  = tile[2][ly][lx];
    const float pyxm = tile[2][ly][lx - 1], pyxp = tile[2][ly][lx + 1];
    const float pyym = tile[2][ly - 1][lx], pyyp = tile[2][ly + 1][lx];

    // roll(x,+1)[i] = x[i-1]; central_diff = (x[i-1] - x[i+1]) * 0.5
    const float lap_r  = rxm + rxp + rym + ryp - 4.0f * rc;
    const float lap_px = pxxm + pxxp + pxym + pxyp - 4.0f * pxc;
    const float lap_py = pyxm + pyxp + pyym + pyyp - 4.0f * pyc;

    const float div_P  = 0.5f * (pxxm - pxxp) + 0.5f * (pyym - pyyp);
    const float dt_rho = -v0 * div_P + Drho * lap_r;

    const float grx   = 0.5f * (rxm - rxp);
    const float gry   = 0.5f * (rym - ryp);
    const float pm2   = pxc * pxc + pyc * pyc;
    const float coeff = -(alpha + beta * pm2);
    const float dt_px = coeff * pxc - v0 * 0.5f * grx + DP * lap_px;
    const float dt_py = coeff * pyc - v0 * 0.5f * gry + DP * lap_py;

    float pred_r  = rc  + DT_STEP * dt_rho;
    const float pred_px = pxc + DT_STEP * dt_px;
    const float pred_py = pyc + DT_STEP * dt_py;

    // softplus(10*x)/10, numerically stable
    {
        const float z = 10.0f * pred_r;
        const float sp = (z > 0.0f) ? (z + log1pf(expf(-z))) : log1pf(expf(z));
        pred_r = sp * 0.1f;
    }

    // Non-temporal stores: outputs are never re-read; keep the 100 MB input
    // resident in the 192 MB L2 so halo re-reads never touch HBM.
    const size_t off = (size_t)b * batch + (size_t)gy * NX + gx;
    __builtin_nontemporal_store(pred_r,  out_pred + off);
    __builtin_nontemporal_store(pred_px, out_pred + off + plane);
    __builtin_nontemporal_store(pred_py, out_pred + off + 2 * plane);
    __builtin_nontemporal_store(dt_rho,  out_dt + off);
    __builtin_nontemporal_store(dt_px,   out_dt + off + plane);
    __builtin_nontemporal_store(dt_py,   out_dt + off + 2 * plane);
}

extern "C" void kernel_launch(void* const* d_in, const int* in_sizes, int n_in,
                              void* d_out, int out_size, void* d_ws, size_t ws_size,
                              hipStream_t stream) {
    (void)in_sizes; (void)n_in; (void)out_size; (void)d_ws; (void)ws_size;
    const float* state  = (const float*)d_in[0];
    const float* v0     = (const float*)d_in[1];
    const float* Drho   = (const float*)d_in[2];
    const float* alpha  = (const float*)d_in[3];
    const float* beta   = (const float*)d_in[4];
    /* d_in[5] = lam : unused by the reference */
    const float* DP     = (const float*)d_in[6];

    float* out_pred = (float*)d_out;
    float* out_dt   = out_pred + (size_t)32 * NC * NX * NY;

    dim3 grid(NX / TX, NY / TY, 32);   // 8 x 64 x 32 blocks
    toner_tu_step_kernel<<<grid, BLOCK, 0, stream>>>(
        state, v0, Drho, alpha, beta, DP, out_pred, out_dt);
}